// OT_Loss_52484500357393
// MI455X (gfx1250) — compile-verified
//
#include <hip/hip_runtime.h>

// ============================================================================
// OT / Sinkhorn loss for MI455X (gfx1250), fp32 WMMA (V_WMMA_F32_16X16X4_F32).
//
// Exploits separability of the squared-distance kernel:
//   K[i, jy*G+jx] = Kx[i,jx] * Ky[i,jy],  Kx,Ky: [1024 x 128]
// so both Sinkhorn matvecs become 128/1024-sized fp32 GEMMs (27 GFLOP total)
// instead of streaming a 268 MB dense K 200 times (~54 GB of HBM traffic).
// One 512-thread workgroup (16 waves) per image runs all 100 iterations with
// barrier-separated phases; V/Vt/u live in LDS (140 KB), Kx/Ky/KyT in
// L2-resident workspace.  512 threads (8 waves/SIMD in CU mode) gives each
// wave enough VGPRs to hold the WMMA accumulators + pointers without the
// scratch spills observed at 1024 threads/workgroup.
//
// Workspace layout (floats): per image Kx[N*G], Ky[N*G], KyT[G*N]; then
// kImgs per-image loss partials.  Needs 4*3*131072*4 B + 16 B ~= 6.3 MB.
// Deterministic: no atomics; fixed-order reductions everywhere.
// ============================================================================

namespace otk {
constexpr int   kG       = 128;        // density grid side
constexpr int   kGG      = kG * kG;    // 16384 cells
constexpr int   kN       = 1024;       // points per image
constexpr int   kImgs    = 4;          // batch
constexpr int   kIter    = 100;        // Sinkhorn iterations
constexpr float kInvReg  = 0.1f;       // 1/REG
constexpr float kEps     = 1e-16f;
constexpr int   kThreads = 512;        // 16 waves (wave32)
constexpr int   kWaves   = kThreads / 32;
constexpr int   kTilesPW = 64 / kWaves; // 16x16 output tiles per wave (= 4)
}

typedef __attribute__((ext_vector_type(2))) float v2f;
typedef __attribute__((ext_vector_type(8))) float v8f;

__device__ __forceinline__ v8f wmma16x16x4(v2f a, v2f b, v8f c) {
  // 8 args: (neg_a, A, neg_b, B, c_mod, C, reuse_a, reuse_b)
  return __builtin_amdgcn_wmma_f32_16x16x4_f32(false, a, false, b, (short)0, c,
                                               false, false);
}

__global__ __launch_bounds__(otk::kThreads, 1)
void ot_sinkhorn_kernel(const float* __restrict__ dens,    // [B,1,G,G]
                        const float* __restrict__ points,  // [B,N,2]
                        float* __restrict__ ws) {
  using namespace otk;
  const int img  = blockIdx.x;
  const int tid  = threadIdx.x;
  const int lane = tid & 31;
  const int wave = tid >> 5;
  const int hg   = lane >> 4;   // half-group: 0 = lanes 0-15, 1 = lanes 16-31
  const int l16  = lane & 15;

  float* Kx  = ws + (size_t)img * (3u * kN * kG);   // [N][G]
  float* Ky  = Kx + kN * kG;                        // [N][G]
  float* KyT = Ky + kN * kG;                        // [G][N]
  float* imgPartials = ws + (size_t)kImgs * (3u * kN * kG);

  __shared__ __align__(16) float sV[kGG];    // V[jy][jx] (also holds M1 = u@K)
  __shared__ __align__(16) float sVT[kGG];   // V transposed [jx][jy]
  __shared__ __align__(16) float sU[kN];     // Sinkhorn u
  __shared__ __align__(16) float sPx[kN];
  __shared__ __align__(16) float sPy[kN];
  __shared__ float sPartial[kWaves];

  // ---- load points, init u = 1/N ------------------------------------------
  for (int i = tid; i < kN; i += kThreads) {
    sPx[i] = points[(size_t)img * kN * 2 + 2 * i + 0];
    sPy[i] = points[(size_t)img * kN * 2 + 2 * i + 1];
    sU[i]  = 1.0f / kN;
  }
  __syncthreads();

  // ---- precompute separable kernel factors Kx, Ky (and KyT) ---------------
  for (int e = tid; e < kN * kG; e += kThreads) {
    const int i = e >> 7;          // point index
    const int j = e & (kG - 1);    // grid coordinate index
    const float c  = (float)j * 8.0f + 4.0f;   // cell-center coordinate
    const float dx = sPx[i] - c;
    const float dy = sPy[i] - c;
    const float kx = __expf(-dx * dx * kInvReg);
    const float ky = __expf(-dy * dy * kInvReg);
    Kx[e] = kx;
    Ky[e] = ky;
    KyT[(size_t)j * kN + i] = ky;
  }
  __threadfence();     // make global Kx/Ky/KyT visible to the whole workgroup
  __syncthreads();

  const float* dimg = dens + (size_t)img * kGG;

  // ---- 100 Sinkhorn iterations --------------------------------------------
  for (int it = 0; it < kIter; ++it) {
    // GEMM1: M1[jy][jx] = sum_k u[k] * KyT[jy][k] * Kx[k][jx]   (= u @ K)
    // 64 output tiles of 16x16, kTilesPW tiles per wave, 256-WMMA K-loop.
    for (int s = 0; s < kTilesPW; ++s) {
      const int t   = wave * kTilesPW + s;
      const int jy0 = (t >> 3) * 16;
      const int jx0 = (t & 7) * 16;
      const float2* arow =
          reinterpret_cast<const float2*>(KyT + (size_t)(jy0 + l16) * kN);
      const float2* u2 = reinterpret_cast<const float2*>(sU);
      v8f acc = {};
      #pragma unroll 2
      for (int k0 = 0; k0 < kN; k0 += 4) {
        const int kp = (k0 >> 1) + hg;       // float2 index: k = k0 + 2*hg
        const float2 av = arow[kp];
        const float2 uv = u2[kp];
        v2f a, b;
        a.x = av.x * uv.x;                   // fold diag(u) into A
        a.y = av.y * uv.y;
        const int ka = k0 + 2 * hg;
        b.x = Kx[(size_t)ka * kG + jx0 + l16];
        b.y = Kx[(size_t)(ka + 1) * kG + jx0 + l16];
        acc = wmma16x16x4(a, b, acc);
      }
      #pragma unroll
      for (int r = 0; r < 8; ++r) {          // C/D layout: M = r + 8*hg, N = l16
        sV[(jy0 + r + 8 * hg) * kG + jx0 + l16] = acc[r];
      }
    }
    __syncthreads();

    // v-update: V = b / (M1 + eps); keep both V and V^T in LDS.
    for (int e = tid; e < kGG; e += kThreads) {
      const float m1 = sV[e];
      const float v  = dimg[e] / (m1 + kEps);
      sV[e] = v;
      sVT[(e & (kG - 1)) * kG + (e >> 7)] = v;
    }
    __syncthreads();

    // GEMM2: T[i][jy] = sum_jx Kx[i][jx] * VT[jx][jy]   (= K @ v, factored)
    // then Kv[i] = sum_jy Ky[i][jy] * T[i][jy]; u = (1/N)/(Kv+eps).
    // Each wave owns kTilesPW 16-row bands -> row reduction stays intra-wave.
    for (int s = 0; s < kTilesPW; ++s) {
      const int i0 = (wave * kTilesPW + s) * 16;
      const float2* arow =
          reinterpret_cast<const float2*>(Kx + (size_t)(i0 + l16) * kG);
      float pk[8];
      #pragma unroll
      for (int r = 0; r < 8; ++r) pk[r] = 0.0f;
      for (int ct = 0; ct < 8; ++ct) {
        const int jy0 = ct * 16;
        v8f acc = {};
        #pragma unroll 2
        for (int k0 = 0; k0 < kG; k0 += 4) {
          const float2 av = arow[(k0 >> 1) + hg];
          const int ka = k0 + 2 * hg;
          v2f a, b;
          a.x = av.x; a.y = av.y;
          b.x = sVT[ka * kG + jy0 + l16];
          b.y = sVT[(ka + 1) * kG + jy0 + l16];
          acc = wmma16x16x4(a, b, acc);
        }
        #pragma unroll
        for (int r = 0; r < 8; ++r) {
          pk[r] += acc[r] * Ky[(size_t)(i0 + r + 8 * hg) * kG + jy0 + l16];
        }
      }
      #pragma unroll
      for (int r = 0; r < 8; ++r) {
        float kv = pk[r];                    // reduce over N lanes (l16 group)
        kv += __shfl_xor(kv, 1, 32);
        kv += __shfl_xor(kv, 2, 32);
        kv += __shfl_xor(kv, 4, 32);
        kv += __shfl_xor(kv, 8, 32);
        if (l16 == 0) sU[i0 + r + 8 * hg] = (1.0f / kN) / (kv + kEps);
      }
    }
    __syncthreads();
  }

  // ---- final loss: sum dis * (u (x) K (x) v), split by separability -------
  //   Term2 uses T = Kx @ V^T, Term1 uses S = Ky @ V; both fused here.
  float lsum = 0.0f;
  for (int s = 0; s < kTilesPW; ++s) {
    const int i0 = (wave * kTilesPW + s) * 16;
    const float2* arowX =
        reinterpret_cast<const float2*>(Kx + (size_t)(i0 + l16) * kG);
    const float2* arowY =
        reinterpret_cast<const float2*>(Ky + (size_t)(i0 + l16) * kG);
    for (int ct = 0; ct < 8; ++ct) {
      const int n0 = ct * 16;
      v8f accT = {};
      v8f accS = {};
      #pragma unroll 2
      for (int k0 = 0; k0 < kG; k0 += 4) {
        const float2 axv = arowX[(k0 >> 1) + hg];
        const float2 ayv = arowY[(k0 >> 1) + hg];
        const int ka = k0 + 2 * hg;
        v2f ax, ay, bT, bS;
        ax.x = axv.x; ax.y = axv.y;
        ay.x = ayv.x; ay.y = ayv.y;
        bT.x = sVT[ka * kG + n0 + l16];
        bT.y = sVT[(ka + 1) * kG + n0 + l16];
        bS.x = sV[ka * kG + n0 + l16];
        bS.y = sV[(ka + 1) * kG + n0 + l16];
        accT = wmma16x16x4(ax, bT, accT);    // T[i][jy]
        accS = wmma16x16x4(ay, bS, accS);    // S[i][jx]
      }
      const float cc = (float)(n0 + l16) * 8.0f + 4.0f;
      #pragma unroll
      for (int r = 0; r < 8; ++r) {
        const int i     = i0 + r + 8 * hg;
        const float ui  = sU[i];
        const float dyv = sPy[i] - cc;       // n is jy for the T term
        const float dxv = sPx[i] - cc;       // n is jx for the S term
        const float kyv = Ky[(size_t)i * kG + n0 + l16];
        const float kxv = Kx[(size_t)i * kG + n0 + l16];
        lsum += ui * (kyv * dyv * dyv * accT[r] + kxv * dxv * dxv * accS[r]);
      }
    }
  }
  #pragma unroll
  for (int off = 16; off >= 1; off >>= 1) lsum += __shfl_xor(lsum, off, 32);
  if (lane == 0) sPartial[wave] = lsum;
  __syncthreads();
  if (tid == 0) {                            // deterministic serial reduction
    float total = 0.0f;
    for (int w = 0; w < kWaves; ++w) total += sPartial[w];
    imgPartials[img] = total;
  }
}

__global__ void ot_finalize_kernel(const float* __restrict__ partials,
                                   float* __restrict__ out) {
  if (threadIdx.x == 0 && blockIdx.x == 0) {
    float s = 0.0f;
    for (int i = 0; i < otk::kImgs; ++i) s += partials[i];
    out[0] = s;     // loss[0]
    out[1] = s;     // wd
    out[2] = 0.0f;  // ot_obj_values[0]
  }
}

extern "C" void kernel_launch(void* const* d_in, const int* in_sizes, int n_in,
                              void* d_out, int out_size, void* d_ws,
                              size_t ws_size, hipStream_t stream) {
  using namespace otk;
  const float* dens   = (const float*)d_in[0];  // normed_density [B,1,G,G]
  // d_in[1] (unnormed_density) is unused by the reference math.
  const float* points = (const float*)d_in[2];  // points [B,N,2]
  float* out = (float*)d_out;
  float* ws  = (float*)d_ws;

  ot_sinkhorn_kernel<<<dim3(kImgs), dim3(kThreads), 0, stream>>>(dens, points,
                                                                 ws);
  float* partials = ws + (size_t)kImgs * (3u * kN * kG);
  ot_finalize_kernel<<<dim3(1), dim3(32), 0, stream>>>(partials, out);
}